// Upsample_Layer_nearest_45509473469139
// MI455X (gfx1250) — compile-verified
//
#include <hip/hip_runtime.h>

typedef __attribute__((ext_vector_type(2))) float v2f;
typedef __attribute__((ext_vector_type(8))) float v8f;

#define CIN        320
#define COUT       160
#define SPATIAL    16384              // 16*32*32 source voxels per (b, c)
#define OUT_PLANE  4096               // 64*64
#define OUT_CHVOL  131072             // 32*64*64

// One wave computes a 16(M out-channels) x 16(N source voxels) tile of the
// low-res conv result with V_WMMA_F32_16X16X4_F32, then replicates each
// result to its 2x2x2 upsampled block with float2 non-temporal stores.
__global__ __launch_bounds__(256)
void Upsample_Layer_nearest_45509473469139_kernel(
    const float* __restrict__ x, const float* __restrict__ Wm,
    const float* __restrict__ bias, float* __restrict__ out)
{
    // W tile for this block's M-tile, pair-interleaved so the A fragment
    // (ISA 16x4 f32 layout) is one conflict-free ds_load_b64 per lane.
    __shared__ float ldsW[16 * CIN];   // 20 KB

    const int tid = threadIdx.x;
    const int m0  = blockIdx.y * 16;   // out-channel tile base (10 tiles)

    for (int i = tid; i < 16 * CIN; i += 256) {
        int m = i / CIN;
        int k = i - m * CIN;
        ldsW[(k >> 1) * 32 + m * 2 + (k & 1)] = Wm[(m0 + m) * CIN + k];
    }
    __syncthreads();

    const int lane  = tid & 31;
    const int wave  = tid >> 5;        // 0..7
    const int half  = lane >> 4;       // 0 or 1
    const int mlane = lane & 15;

    // This lane's GEMM column = one source voxel (flattened over batch+space)
    const int n    = (blockIdx.x * 8 + wave) * 16 + mlane;
    const int bidx = n >> 14;                  // batch
    const int spat = n & (SPATIAL - 1);        // d*1024 + h*32 + w

    const float* xcol = x + (size_t)bidx * CIN * SPATIAL + spat;

    v8f acc = {0.f, 0.f, 0.f, 0.f, 0.f, 0.f, 0.f, 0.f};

#pragma unroll 4
    for (int k = 0; k < CIN; k += 4) {
        const int kb = k + 2 * half;   // lanes 16-31 handle K+2,K+3 (ISA layout)
        v2f a = *(const v2f*)&ldsW[(kb >> 1) * 32 + mlane * 2];
        v2f b;
        b.x = xcol[(size_t)kb * SPATIAL];
        b.y = xcol[(size_t)(kb + 1) * SPATIAL];
        acc = __builtin_amdgcn_wmma_f32_16x16x4_f32(
            /*neg_a=*/false, a, /*neg_b=*/false, b,
            /*c_mod=*/(short)0, acc, /*reuse_a=*/false, /*reuse_b=*/false);
    }

    // C/D layout: VGPR r, lanes 0-15 -> M=r, lanes 16-31 -> M=r+8
    const int mrow0 = m0 + 8 * half;
    float bv[8];
#pragma unroll
    for (int r = 0; r < 8; ++r) bv[r] = bias[mrow0 + r];

    const int d = spat >> 10;
    const int h = (spat >> 5) & 31;
    const int w = spat & 31;
    float* obase = out + (size_t)bidx * COUT * OUT_CHVOL
                       + (size_t)(2 * d) * OUT_PLANE + (2 * h) * 64 + 2 * w;

#pragma unroll
    for (int r = 0; r < 8; ++r) {
        const int o  = mrow0 + r;
        float val    = acc[r] + bv[r];
        v2f v        = {val, val};
        float* p     = obase + (size_t)o * OUT_CHVOL;
        // 2x2x2 replication; w-pair is contiguous -> 128B/16-lane b64 stores.
        // Non-temporal: keep the 168 MB output stream from evicting x in L2.
        __builtin_nontemporal_store(v, (v2f*)p);
        __builtin_nontemporal_store(v, (v2f*)(p + 64));
        __builtin_nontemporal_store(v, (v2f*)(p + OUT_PLANE));
        __builtin_nontemporal_store(v, (v2f*)(p + OUT_PLANE + 64));
    }
}

extern "C" void kernel_launch(void* const* d_in, const int* in_sizes, int n_in,
                              void* d_out, int out_size, void* d_ws, size_t ws_size,
                              hipStream_t stream) {
    const float* x    = (const float*)d_in[0];   // [2,320,16,32,32]
    const float* Wm   = (const float*)d_in[1];   // [160,320]
    const float* bias = (const float*)d_in[2];   // [160]
    float* out        = (float*)d_out;           // [2,160,32,64,64]

    dim3 grid(32768 / (8 * 16), COUT / 16);      // (256, 10)
    dim3 block(256);
    Upsample_Layer_nearest_45509473469139_kernel<<<grid, block, 0, stream>>>(
        x, Wm, bias, out);
}